// NewExchange_20220706030376
// MI455X (gfx1250) — compile-verified
//
#include <hip/hip_runtime.h>
#include <hip/hip_bf16.h>
#include <stdint.h>

// Problem shape: x1,x2 are (B=32, C=256, H=56, W=56) fp32.
// Every output plane (b,c) is a verbatim copy of one input plane, chosen by
// per-channel metadata derived from w1/w2/threshold. Pure HBM-bound copy:
// ~411 MB total traffic -> ~17.6us floor at 23.3 TB/s.

#define NCH        256
#define PLANE      3136           // 56*56 floats
#define PLANE_B    12544          // bytes per plane
#define PLANES_TOT 16384          // 2 outputs * 32 batches * 256 channels
#define WAVES_PER_BLOCK 4

// ---------------------------------------------------------------------------
// Kernel 1: per-channel metadata (tiny: 256 channels).
// Reproduces jnp.argsort(-|w|) stable-descending semantics via all-pairs rank,
// cumsum of below-threshold flags, and the cross-gather order2[rank1] /
// order1[rank2]. Emits a 512-entry routing table:
//   table[c]        (feature1, channel c): src_channel | (from_x2 ? 256 : 0)
//   table[256 + c]  (feature2, channel c): src_channel | (from_x2 ? 256 : 0)
// ---------------------------------------------------------------------------
__global__ __launch_bounds__(NCH) void meta_kernel(const float* __restrict__ w1,
                                                   const float* __restrict__ w2,
                                                   const float* __restrict__ thr_p,
                                                   int* __restrict__ table) {
    __shared__ float a1[NCH], a2[NCH];
    __shared__ int   order1[NCH], order2[NCH];
    __shared__ unsigned char bel1[NCH], bel2[NCH];

    const int c = threadIdx.x;
    a1[c] = fabsf(w1[c]);
    a2[c] = fabsf(w2[c]);
    __syncthreads();

    const float v1 = a1[c];
    const float v2 = a2[c];
    int p1 = 0, p2 = 0;
    // stable descending rank: elements strictly larger, or equal with lower idx
    for (int j = 0; j < NCH; ++j) {
        const float b1 = a1[j], b2 = a2[j];
        p1 += (b1 > v1) || ((b1 == v1) && (j < c));
        p2 += (b2 > v2) || ((b2 == v2) && (j < c));
    }
    order1[p1] = c;
    order2[p2] = c;

    const float thr = thr_p[0];
    bel1[c] = (v1 < thr) ? 1 : 0;
    bel2[c] = (v2 < thr) ? 1 : 0;
    __syncthreads();

    // inclusive cumsum of below flags, then rank = clip(cumsum-1, 0)
    int r1 = 0, r2 = 0;
    for (int j = 0; j <= c; ++j) { r1 += bel1[j]; r2 += bel2[j]; }
    r1 = (r1 > 0) ? (r1 - 1) : 0;
    r2 = (r2 > 0) ? (r2 - 1) : 0;

    // feature1: below1 -> pull x2[:, order2[rank1]]   else x1[:, c]
    // feature2: below2 -> pull x1[:, order1[rank2]]   else x2[:, c]
    const int t1 = bel1[c] ? (order2[r1] | 256) : c;
    const int t2 = bel2[c] ? (order1[r2])       : (c | 256);
    table[c]        = t1;
    table[NCH + c]  = t2;
}

// ---------------------------------------------------------------------------
// Kernel 2: bulk plane copy via the CDNA5 async global<->LDS engine.
// One wave per 12,544-byte plane. Data never touches VGPRs:
//   25 async loads (24 x B128 + 1 x B64) global -> LDS slice
//   s_wait_asynccnt 0
//   25 async stores LDS slice -> global
// Each wave keeps 25 asyncs (12.5 KB) in flight; 4 waves/block, 4096 blocks.
// ---------------------------------------------------------------------------
__global__ __launch_bounds__(WAVES_PER_BLOCK * 32) void copy_kernel(
        const float* __restrict__ x1, const float* __restrict__ x2,
        const int* __restrict__ table, float* __restrict__ out) {
    __shared__ __align__(16) float lbuf[WAVES_PER_BLOCK * PLANE]; // 50,176 B

    const int wave = threadIdx.x >> 5;
    const int lane = threadIdx.x & 31;
    const int plane = blockIdx.x * WAVES_PER_BLOCK + wave; // [0, 16384)

    const int which = plane >> 13;      // 0 = feature1, 1 = feature2
    const int bc    = plane & 8191;     // b*256 + c
    const int c     = bc & 255;

    int t = table[(which << 8) | c];
    t = __builtin_amdgcn_readfirstlane(t);   // wave-uniform routing entry

    const float* srcT = (t & 256) ? x2 : x1;
    const unsigned long long srcBase =
        (unsigned long long)(uintptr_t)(srcT + (size_t)((bc & ~255) | (t & 255)) * PLANE);
    const unsigned long long dstBase =
        (unsigned long long)(uintptr_t)(out + (size_t)plane * PLANE);

    // LDS byte offset of this wave's slice (generic-ptr low 32 bits == LDS addr)
    const unsigned ldsBase = (unsigned)(uintptr_t)(&lbuf[wave * PLANE]);

    // ---- async loads: global -> LDS ----
    {
        unsigned voff = (unsigned)(lane * 16);
        unsigned lds  = ldsBase + voff;
#pragma unroll
        for (int k = 0; k < 24; ++k) {
            asm volatile("global_load_async_to_lds_b128 %0, %1, %2"
                         :: "v"(lds), "v"(voff), "s"(srcBase) : "memory");
            voff += 512;
            lds  += 512;
        }
        unsigned vt = 12288u + (unsigned)(lane * 8);  // 256-byte tail
        unsigned lt = ldsBase + vt;
        asm volatile("global_load_async_to_lds_b64 %0, %1, %2"
                     :: "v"(lt), "v"(vt), "s"(srcBase) : "memory");
    }

    asm volatile("s_wait_asynccnt 0" ::: "memory");

    // ---- async stores: LDS -> global ----
    {
        unsigned voff = (unsigned)(lane * 16);
        unsigned lds  = ldsBase + voff;
#pragma unroll
        for (int k = 0; k < 24; ++k) {
            asm volatile("global_store_async_from_lds_b128 %0, %1, %2"
                         :: "v"(voff), "v"(lds), "s"(dstBase) : "memory");
            voff += 512;
            lds  += 512;
        }
        unsigned vt = 12288u + (unsigned)(lane * 8);
        unsigned lt = ldsBase + vt;
        asm volatile("global_store_async_from_lds_b64 %0, %1, %2"
                     :: "v"(vt), "v"(lt), "s"(dstBase) : "memory");
    }

    asm volatile("s_wait_asynccnt 0" ::: "memory");
}

// ---------------------------------------------------------------------------
extern "C" void kernel_launch(void* const* d_in, const int* in_sizes, int n_in,
                              void* d_out, int out_size, void* d_ws, size_t ws_size,
                              hipStream_t stream) {
    const float* x1  = (const float*)d_in[0];
    const float* x2  = (const float*)d_in[1];
    const float* w1  = (const float*)d_in[2];
    const float* w2  = (const float*)d_in[3];
    const float* thr = (const float*)d_in[4];
    float* out = (float*)d_out;
    int* table = (int*)d_ws;   // 512 ints of scratch

    meta_kernel<<<1, NCH, 0, stream>>>(w1, w2, thr, table);

    const int blocks = PLANES_TOT / WAVES_PER_BLOCK;  // 4096
    copy_kernel<<<blocks, WAVES_PER_BLOCK * 32, 0, stream>>>(x1, x2, table, out);
}